// DynamicMaskedGATv2Layer_42039139893268
// MI455X (gfx1250) — compile-verified
//
#include <hip/hip_runtime.h>
#include <hip/hip_bf16.h>

typedef float v2f __attribute__((ext_vector_type(2)));
typedef float v8f __attribute__((ext_vector_type(8)));

#define LRS 68           // LDS row stride in floats: 272B -> 16B aligned rows, bank = (4*row + col) % 64
#define NEG_INF_F (-1e9f)

__launch_bounds__(256)
__global__ void gatv2_kernel(const float* __restrict__ feat,   // [BT=256, 64, 64]
                             const int*   __restrict__ adj,    // [8, 64, 64]
                             const float* __restrict__ Wsrc,   // [4, 64, 64]
                             const float* __restrict__ Wdst,   // [4, 64, 64]
                             const float* __restrict__ avec,   // [4, 64]
                             float*       __restrict__ out)    // [BT=256, 64, 64]
{
    __shared__ float s_feat[64 * LRS];
    __shared__ float s_wa  [64 * LRS];
    __shared__ float s_wb  [64 * LRS];
    __shared__ float s_hs  [64 * LRS];
    __shared__ float s_hd  [64 * LRS];
    __shared__ float s_at  [64 * LRS];
    __shared__ float s_a   [64];
    __shared__ unsigned s_mask[64][2];

    const int bt   = blockIdx.x;        // 0..255  (b*T + t)
    const int g    = bt >> 5;           // adjacency graph index = bt / T (T=32)
    const int t    = threadIdx.x;       // 0..255
    const int lane = t & 31;
    const int wave = t >> 5;            // 0..7

    // WMMA f32 16x16x4 fragment coordinates (wave32):
    const int fm = lane & 15;           // m (A) / n (B,C,D) inside tile
    const int kb = (lane >> 4) << 1;    // K base: lanes 0-15 -> K{0,1}, lanes 16-31 -> K{2,3}
    const int mh = (lane >> 4) << 3;    // C/D row offset: +0 or +8

    // ---- stage features for this graph (float4, 16B aligned both sides) ----
    const float* fsrc = feat + (size_t)bt * 64 * 64;
    for (int idx = t; idx < 64 * 16; idx += 256) {
        const int r = idx >> 4, c4 = (idx & 15) << 2;
        *(float4*)(&s_feat[r * LRS + c4]) = *(const float4*)(fsrc + r * 64 + c4);
    }
    // ---- adjacency -> bitmask (self loops forced on) ----
    if (t < 128) {
        const int r = t >> 1, half = t & 1;
        const int* arow = adj + ((size_t)g * 64 + r) * 64 + half * 32;
        unsigned bits = 0u;
        for (int j = 0; j < 32; ++j) bits |= (arow[j] != 0) ? (1u << j) : 0u;
        const int dj = r - half * 32;
        if (dj >= 0 && dj < 32) bits |= (1u << dj);
        s_mask[r][half] = bits;
    }

    v8f accA = {};   // persistent head-mean accumulators: tiles 2*wave, 2*wave+1
    v8f accB = {};

    for (int h = 0; h < 4; ++h) {
        __syncthreads();   // feat/mask ready (h==0); previous head's GEMM3 reads done (h>0)

        // ---- stage this head's weights + attention vector ----
        const float* wsp = Wsrc + h * 64 * 64;
        const float* wdp = Wdst + h * 64 * 64;
        for (int idx = t; idx < 64 * 16; idx += 256) {
            const int r = idx >> 4, c4 = (idx & 15) << 2;
            *(float4*)(&s_wa[r * LRS + c4]) = *(const float4*)(wsp + r * 64 + c4);
            *(float4*)(&s_wb[r * LRS + c4]) = *(const float4*)(wdp + r * 64 + c4);
        }
        if (t < 64) s_a[t] = avec[h * 64 + t];
        if (h < 3 && t < 64) {  // pull next head's weights toward the caches
            __builtin_prefetch(wsp + 64 * 64 + t * 64, 0, 1);
            __builtin_prefetch(wdp + 64 * 64 + t * 64, 0, 1);
        }
        __syncthreads();

        // ---- GEMM1/2: h_src = feat@Wsrc, h_dst = feat@Wdst (32 tile jobs, 4 per wave) ----
        for (int job = wave * 4; job < wave * 4 + 4; ++job) {
            const int which = job >> 4;                  // 0 -> h_src, 1 -> h_dst
            const int tile  = job & 15;
            const int ti = (tile >> 2) << 4, tj = (tile & 3) << 4;
            const float* Bm  = which ? s_wb : s_wa;
            float*       Dst = which ? s_hd : s_hs;
            v8f c = {};
            #pragma unroll
            for (int k0 = 0; k0 < 64; k0 += 4) {
                v2f av = *(const v2f*)(&s_feat[(ti + fm) * LRS + k0 + kb]);
                v2f bv;
                bv.x = Bm[(k0 + kb    ) * LRS + tj + fm];
                bv.y = Bm[(k0 + kb + 1) * LRS + tj + fm];
                c = __builtin_amdgcn_wmma_f32_16x16x4_f32(false, av, false, bv,
                                                          (short)0, c, false, false);
            }
            #pragma unroll
            for (int v = 0; v < 8; ++v)
                Dst[(ti + v + mh) * LRS + tj + fm] = c[v];
        }
        __syncthreads();

        // ---- pairwise GATv2 scores + masked softmax ----
        {
            const int i = t >> 2;          // node row
            const int q = t & 3;           // owns j = q, q+4, ..., q+60
            float acc[16];
            #pragma unroll
            for (int jj = 0; jj < 16; ++jj) acc[jj] = 0.f;

            for (int u = 0; u < 64; ++u) {
                const float hsv = s_hs[i * LRS + u];
                const float au  = s_a[u];
                #pragma unroll
                for (int jj = 0; jj < 16; ++jj) {
                    float x = hsv + s_hd[(q + (jj << 2)) * LRS + u];
                    x = fmaxf(x, 0.2f * x);               // leaky_relu(., 0.2)
                    acc[jj] = fmaf(x, au, acc[jj]);
                }
            }
            const unsigned m0 = s_mask[i][0], m1 = s_mask[i][1];
            float mx = -3.0e38f;
            #pragma unroll
            for (int jj = 0; jj < 16; ++jj) {
                const int j = q + (jj << 2);
                const bool on = (j < 32) ? ((m0 >> j) & 1u) : ((m1 >> (j - 32)) & 1u);
                acc[jj] = on ? acc[jj] : NEG_INF_F;
                mx = fmaxf(mx, acc[jj]);
            }
            mx = fmaxf(mx, __shfl_xor(mx, 1, 32));        // row spans 4 adjacent lanes
            mx = fmaxf(mx, __shfl_xor(mx, 2, 32));
            float sum = 0.f;
            #pragma unroll
            for (int jj = 0; jj < 16; ++jj) { acc[jj] = __expf(acc[jj] - mx); sum += acc[jj]; }
            sum += __shfl_xor(sum, 1, 32);
            sum += __shfl_xor(sum, 2, 32);
            const float inv = 1.0f / sum;
            #pragma unroll
            for (int jj = 0; jj < 16; ++jj)
                s_at[i * LRS + q + (jj << 2)] = acc[jj] * inv;
        }
        __syncthreads();

        // ---- GEMM3: out += attn @ h_src (2 tiles per wave, accumulated across heads) ----
        #pragma unroll
        for (int s = 0; s < 2; ++s) {
            const int tile = wave * 2 + s;
            const int ti = (tile >> 2) << 4, tj = (tile & 3) << 4;
            v8f c = s ? accB : accA;
            #pragma unroll
            for (int k0 = 0; k0 < 64; k0 += 4) {
                v2f av = *(const v2f*)(&s_at[(ti + fm) * LRS + k0 + kb]);
                v2f bv;
                bv.x = s_hs[(k0 + kb    ) * LRS + tj + fm];
                bv.y = s_hs[(k0 + kb + 1) * LRS + tj + fm];
                c = __builtin_amdgcn_wmma_f32_16x16x4_f32(false, av, false, bv,
                                                          (short)0, c, false, false);
            }
            if (s) accB = c; else accA = c;
        }
    }

    // ---- mean over heads, store (each element written exactly once) ----
    float* op = out + (size_t)bt * 64 * 64;
    #pragma unroll
    for (int s = 0; s < 2; ++s) {
        const int tile = wave * 2 + s;
        const int ti = (tile >> 2) << 4, tj = (tile & 3) << 4;
        const v8f c = s ? accB : accA;
        #pragma unroll
        for (int v = 0; v < 8; ++v)
            op[(ti + v + mh) * 64 + tj + fm] = 0.25f * c[v];
    }
}

extern "C" void kernel_launch(void* const* d_in, const int* in_sizes, int n_in,
                              void* d_out, int out_size, void* d_ws, size_t ws_size,
                              hipStream_t stream) {
    const float* feat = (const float*)d_in[0];   // [8,32,64,64]
    const int*   adj  = (const int*)  d_in[1];   // [8,64,64]
    const float* wsrc = (const float*)d_in[2];   // [4,64,64]
    const float* wdst = (const float*)d_in[3];   // [4,64,64]
    const float* a    = (const float*)d_in[4];   // [4,64,1]
    float*       out  = (float*)d_out;           // [8,32,64,64]
    (void)in_sizes; (void)n_in; (void)out_size; (void)d_ws; (void)ws_size;

    gatv2_kernel<<<256, 256, 0, stream>>>(feat, adj, wsrc, wdst, a, out);
}